// LocalSelfAttention_30648886624829
// MI455X (gfx1250) — compile-verified
//
#include <hip/hip_runtime.h>
#include <cmath>

// CDNA5 fp32 WMMA fragment types: 16x16x4 f32 -> A/B = 2 VGPRs, C/D = 8 VGPRs.
typedef float v2f __attribute__((ext_vector_type(2)));
typedef float v8f __attribute__((ext_vector_type(8)));
typedef unsigned int u32x4 __attribute__((ext_vector_type(4)));
typedef int i32x4 __attribute__((ext_vector_type(4)));
typedef int i32x8 __attribute__((ext_vector_type(8)));

#if defined(__has_builtin)
#if __has_builtin(__builtin_amdgcn_tensor_load_to_lds)
#define HAVE_TDM 1
#endif
#endif
#ifndef HAVE_TDM
#define HAVE_TDM 0
#endif

__device__ __forceinline__ v8f wmma4(v2f a, v2f b, v8f c) {
    // (neg_a, A, neg_b, B, c_mod, C, reuse_a, reuse_b)
    return __builtin_amdgcn_wmma_f32_16x16x4_f32(false, a, false, b, (short)0, c,
                                                 false, false);
}

// Low 32 bits of a generic pointer into the LDS aperture = wave-relative LDS
// byte address (ISA 10.2).
__device__ __forceinline__ unsigned lds_addr_of(const void* p) {
    return (unsigned)(unsigned long long)p;
}

#if HAVE_TDM
// Tensor Data Mover: copy a 2D tile (tile_w x tile_h elements of f32, row
// stride stride_elems) from global memory into LDS at lds_byte, with HW
// padding inserted after each pad interval (bank-conflict avoidance).
// D# layout per CDNA5 ISA; groups 2/3 zero (2D).
// 6-arg builtin form (clang-23 / therock-10.0 headers).
__device__ __forceinline__ void tdm_load_2d(unsigned lds_byte, const float* gptr,
                                            int tensor_w, int tensor_h,
                                            int tile_w, int tile_h,
                                            int stride_elems,
                                            int pad_interval_code,
                                            int pad_amount_code) {
    unsigned long long ga = (unsigned long long)gptr;
    u32x4 g0;
    g0.x = 1u;                                   // count=1 (valid descriptor)
    g0.y = lds_byte;                             // lds_addr
    g0.z = (unsigned)(ga & 0xFFFFFFFFull);       // global_addr[31:0]
    g0.w = (unsigned)((ga >> 32) & 0x1FFFFFFull) // global_addr[56:32]
           | 0x80000000u;                        // type=2 ("image") in [127:126]
    i32x8 g1;
    g1[0] = (2 << 16)                            // data_size = 4 bytes
          | (1 << 20)                            // pad_enable
          | (pad_interval_code << 22)            // pad interval (2^(c+1) dwords)
          | (pad_amount_code << 25);             // pad amount (c+1 dwords)
    g1[1] = (int)((tensor_w & 0xFFFF) << 16);                       // dim0 lo16
    g1[2] = (int)(((unsigned)tensor_w >> 16) | ((tensor_h & 0xFFFF) << 16));
    g1[3] = (int)(((unsigned)tensor_h >> 16) | ((unsigned)tile_w << 16));
    g1[4] = tile_h;                              // tile_dim1; tile_dim2 = 0
    g1[5] = stride_elems;                        // tensor_dim0_stride[31:0]
    g1[6] = 0;                                   // stride0 hi / stride1 lo
    g1[7] = 0;
    i32x4 z4 = {0, 0, 0, 0};
    i32x8 z8 = {0, 0, 0, 0, 0, 0, 0, 0};
    __builtin_amdgcn_tensor_load_to_lds(g0, g1, z4, z4, z8, 0);
}
#endif

// ---------------------------------------------------------------------------
// Tiled fp32 GEMM: C[M][N] = A[M][K] * B[K][N] (+ bias[N] if bias != null)
// Block: 128 threads = 4 wave32. Block tile 64x64, each wave owns 16 rows.
// K staged 16-wide in LDS by the Tensor Data Mover (wave0: A tile, wave1: B
// tile), with HW padding: As rows stride 20 (16 lanes -> 16 distinct banks),
// Bs rows stride 68.
// ---------------------------------------------------------------------------
__global__ __launch_bounds__(128) void gemm_wmma_f32(
    const float* __restrict__ A, const float* __restrict__ B,
    const float* __restrict__ bias, float* __restrict__ C,
    int M, int N, int K) {
    __shared__ float As[64][20];   // [m][k], pad 16->20
    __shared__ float Bs[16][68];   // [k][n], pad 64->68

    const int tid  = threadIdx.x;
    const int wid  = tid >> 5;
    const int lane = tid & 31;
    const int ln   = lane & 15;    // N (or M) index within a 16-wide tile
    const int hh   = lane >> 4;    // half-wave: selects K pair {2hh, 2hh+1}
    const int bM   = blockIdx.y * 64;
    const int bN   = blockIdx.x * 64;
    const int m0   = wid * 16;

    const v8f vz = {0.f, 0.f, 0.f, 0.f, 0.f, 0.f, 0.f, 0.f};
    v8f acc[4] = {vz, vz, vz, vz};

    for (int k0 = 0; k0 < K; k0 += 16) {
        __syncthreads();   // previous iteration's LDS reads done
#if HAVE_TDM
        if (wid == 0)      // A tile: 16 wide x 64 rows, row stride K
            tdm_load_2d(lds_addr_of(&As[0][0]), A + (size_t)bM * K + k0,
                        K, M, 16, 64, K, /*interval 16dw*/ 3, /*pad 4dw*/ 3);
        if (wid == 1)      // B tile: 64 wide x 16 rows, row stride N
            tdm_load_2d(lds_addr_of(&Bs[0][0]), B + (size_t)k0 * N + bN,
                        N, K, 64, 16, N, /*interval 64dw*/ 5, /*pad 4dw*/ 3);
        __builtin_amdgcn_s_wait_tensorcnt(0);
#else
#pragma unroll
        for (int i = 0; i < 2; ++i) {
            int idx4 = tid + i * 128;            // [0,256)
            int row  = idx4 >> 2;                // 4 float4 per A row
            int ca   = (idx4 & 3) << 2;
            float4 fa = *(const float4*)(A + (size_t)(bM + row) * K + k0 + ca);
            *(float4*)&As[row][ca] = fa;
            int kr = idx4 >> 4;                  // 16 float4 per B row
            int cb = (idx4 & 15) << 2;
            float4 fb = *(const float4*)(B + (size_t)(k0 + kr) * N + bN + cb);
            *(float4*)&Bs[kr][cb] = fb;
        }
#endif
        __syncthreads();   // staged tiles visible to all waves

#pragma unroll
        for (int kk = 0; kk < 4; ++kk) {
            v2f a = *(const v2f*)&As[m0 + ln][4 * kk + 2 * hh];
#pragma unroll
            for (int t = 0; t < 4; ++t) {
                v2f bf;
                bf.x = Bs[4 * kk + 2 * hh + 0][t * 16 + ln];
                bf.y = Bs[4 * kk + 2 * hh + 1][t * 16 + ln];
                acc[t] = wmma4(a, bf, acc[t]);
            }
        }
    }

    // C/D layout: VGPR v, lanes 0-15 -> row v, lanes 16-31 -> row v+8.
#pragma unroll
    for (int t = 0; t < 4; ++t) {
        int n = bN + t * 16 + ln;
        float bv = (bias != nullptr) ? bias[n] : 0.0f;
#pragma unroll
        for (int v = 0; v < 8; ++v) {
            int m = bM + m0 + v + 8 * hh;
            C[(size_t)m * N + n] = acc[t][v] + bv;
        }
    }
}

// ---------------------------------------------------------------------------
// Windowed attention. One wave per (b, h, 16-query tile). Band of 80 keys
// [s0-32, s0+48). Scores: 5 tiles x 16 k-steps of WMMA f32 16x16x4.
// Zero-padded OOB keys -> K fragment zeroed (score 0, matches reference's
// zero-pad entering softmax); window mask r<=j<=r+64 -> -1e30 before exp.
// P transposed C-layout -> A-layout via LDS, then P@V: 4 tiles x 20 k-steps.
// ---------------------------------------------------------------------------
#define S_LEN 2048
#define QKV_LD 1536

__global__ __launch_bounds__(128) void attn_wmma_f32(
    const float* __restrict__ qkv, float* __restrict__ attn) {
    __shared__ float Pl[4][16][84];   // [wave][row][band-col], pad 84

    const int tid  = threadIdx.x;
    const int wid  = tid >> 5;
    const int lane = tid & 31;
    const int ln   = lane & 15;
    const int hh   = lane >> 4;

    const int gw   = blockIdx.x * 4 + wid;   // 0..2047
    const int b    = gw >> 10;
    const int rem  = gw & 1023;
    const int h    = rem >> 7;
    const int s0   = (rem & 127) << 4;
    const int qoff = h * 192;                // q | k(+64) | v(+128)

    const v8f vz = {0.f, 0.f, 0.f, 0.f, 0.f, 0.f, 0.f, 0.f};

    // Q fragments for all 16 k-steps (K dim = 64).
    v2f aq[16];
    const float* qrow = qkv + (size_t)(b * S_LEN + s0 + ln) * QKV_LD + qoff;
#pragma unroll
    for (int kk = 0; kk < 16; ++kk)
        aq[kk] = *(const v2f*)(qrow + 4 * kk + 2 * hh);

    // Scores S = Q @ K^T over the 80-wide band.
    float p[5][8];
#pragma unroll
    for (int t = 0; t < 5; ++t) {
        int key = s0 - 32 + t * 16 + ln;          // this lane's key (B-layout N)
        int kc  = min(max(key, 0), S_LEN - 1);
        float msk = (key >= 0 && key < S_LEN) ? 1.0f : 0.0f;
        const float* krow = qkv + (size_t)(b * S_LEN + kc) * QKV_LD + qoff + 64;
        v8f c = vz;
#pragma unroll
        for (int kk = 0; kk < 16; ++kk) {
            v2f bf = *(const v2f*)(krow + 4 * kk + 2 * hh);
            bf.x *= msk; bf.y *= msk;             // zero-padded key -> score 0
            c = wmma4(aq[kk], bf, c);
        }
#pragma unroll
        for (int v = 0; v < 8; ++v) p[t][v] = c[v];
    }

    // Row softmax: rows live per half-wave (r = v + 8*hh), cols across 16 lanes
    // and 5 accumulators -> local combine + 16-lane shfl_xor reductions.
#pragma unroll
    for (int v = 0; v < 8; ++v) {
        int r = v + 8 * hh;
        float mx = -1e30f;
#pragma unroll
        for (int t = 0; t < 5; ++t) {
            int j = t * 16 + ln;
            bool inwin = (j >= r) && (j <= r + 64);
            float s = inwin ? p[t][v] * 0.125f : -1e30f;   // 1/sqrt(64)
            p[t][v] = s;
            mx = fmaxf(mx, s);
        }
#pragma unroll
        for (int d = 1; d < 16; d <<= 1) mx = fmaxf(mx, __shfl_xor(mx, d, 32));
        float sum = 0.f;
#pragma unroll
        for (int t = 0; t < 5; ++t) {
            float e = __expf(p[t][v] - mx);
            p[t][v] = e;
            sum += e;
        }
#pragma unroll
        for (int d = 1; d < 16; d <<= 1) sum += __shfl_xor(sum, d, 32);
        float inv = 1.0f / sum;
#pragma unroll
        for (int t = 0; t < 5; ++t) p[t][v] *= inv;
    }

    // Transpose P (C/D layout -> A layout) through LDS.
#pragma unroll
    for (int t = 0; t < 5; ++t)
#pragma unroll
        for (int v = 0; v < 8; ++v)
            Pl[wid][v + 8 * hh][t * 16 + ln] = p[t][v];
    __syncthreads();

    // O = P @ V  (K dim = 80 band positions -> 20 k-steps).
    v8f o[4] = {vz, vz, vz, vz};
#pragma unroll 4
    for (int kk = 0; kk < 20; ++kk) {
        v2f a = *(const v2f*)&Pl[wid][ln][4 * kk + 2 * hh];
        int k0 = s0 - 32 + 4 * kk + 2 * hh;
        int k1 = k0 + 1;
        int c0 = min(max(k0, 0), S_LEN - 1);
        int c1 = min(max(k1, 0), S_LEN - 1);
        float mk0 = (k0 >= 0 && k0 < S_LEN) ? 1.0f : 0.0f;  // OOB v rows -> 0
        float mk1 = (k1 >= 0 && k1 < S_LEN) ? 1.0f : 0.0f;
        const float* v0 = qkv + (size_t)(b * S_LEN + c0) * QKV_LD + qoff + 128;
        const float* v1 = qkv + (size_t)(b * S_LEN + c1) * QKV_LD + qoff + 128;
#pragma unroll
        for (int t = 0; t < 4; ++t) {
            v2f bf;
            bf.x = v0[t * 16 + ln] * mk0;
            bf.y = v1[t * 16 + ln] * mk1;
            o[t] = wmma4(a, bf, o[t]);
        }
    }

    // Store attn tile into [B*S, 512] at head column h*64.
#pragma unroll
    for (int t = 0; t < 4; ++t)
#pragma unroll
        for (int v = 0; v < 8; ++v) {
            int r = v + 8 * hh;
            attn[(size_t)(b * S_LEN + s0 + r) * 512 + h * 64 + t * 16 + ln] = o[t][v];
        }
}

// ---------------------------------------------------------------------------
extern "C" void kernel_launch(void* const* d_in, const int* in_sizes, int n_in,
                              void* d_out, int out_size, void* d_ws, size_t ws_size,
                              hipStream_t stream) {
    const float* x     = (const float*)d_in[0];   // [2,2048,512]
    const float* W_qkv = (const float*)d_in[1];   // [512,1536]
    const float* W_out = (const float*)d_in[2];   // [512,512]
    const float* b_out = (const float*)d_in[3];   // [512]
    float* out = (float*)d_out;                   // [2,2048,512]

    float* qkv  = (float*)d_ws;                   // 4096*1536 f32 = 24 MiB
    float* attn = qkv + (size_t)4096 * 1536;      // 4096*512  f32 =  8 MiB

    // 1) qkv = x @ W_qkv                (M=4096, N=1536, K=512)
    gemm_wmma_f32<<<dim3(1536 / 64, 4096 / 64), 128, 0, stream>>>(
        x, W_qkv, nullptr, qkv, 4096, 1536, 512);

    // 2) windowed attention             (2048 wave-tiles, 4 waves/block)
    attn_wmma_f32<<<512, 128, 0, stream>>>(qkv, attn);

    // 3) out = attn @ W_out + b_out     (M=4096, N=512, K=512)
    gemm_wmma_f32<<<dim3(512 / 64, 4096 / 64), 128, 0, stream>>>(
        attn, W_out, b_out, out, 4096, 512, 512);
}